// SNN_53996328845399
// MI455X (gfx1250) — compile-verified
//
#include <hip/hip_runtime.h>

typedef __attribute__((ext_vector_type(2))) float v2f;
typedef __attribute__((ext_vector_type(8))) float v8f;

#define VOCAB 25000
#define EMB   100
#define TSTEP 256
#define BATCH 512
#define SENT  500
#define THRES 25.0f

// ---------------------------------------------------------------------------
// Kernel 1: embedding gather + mean pool.  One block per batch element.
// Lane -> embedding dim e, so each gathered table row is a coalesced 400B read
// (table is 10 MB, lives in the 192 MB L2 after first touch).
// ---------------------------------------------------------------------------
__global__ __launch_bounds__(128) void pool_kernel(const int* __restrict__ text,
                                                   const float* __restrict__ emb,
                                                   float* __restrict__ pooled) {
    __shared__ int tbuf[SENT];
    const int b   = blockIdx.x;
    const int tid = threadIdx.x;
    for (int i = tid; i < SENT; i += 128) tbuf[i] = text[b * SENT + i];
    __syncthreads();
    if (tid < EMB) {
        float acc = 0.0f;
        #pragma unroll 4
        for (int s = 0; s < SENT; ++s) {
            acc += emb[(size_t)tbuf[s] * EMB + tid];
        }
        pooled[b * EMB + tid] = acc / (float)SENT;   // mean over sentence
    }
}

// ---------------------------------------------------------------------------
// Kernel 2: fused Bernoulli spike generation + synaptic currents (WMMA) + LIF.
// One block (256 thr = 8 waves) per batch element b.
//
// Currents: I[t] = sum_e fc[e] * (pooled[b,e] > rand[b,e,t])  -- a GEMV done
// on the matrix core with V_WMMA_F32_16X16X4_F32 (f32 in/out, so numerics
// match the f32 reference):
//   * per wave-tile: M = 16 timesteps, K = 4 embedding dims, 25 K-steps (E=100)
//   * A layout (16x4 f32): lanes 0-15 carry K=0,1; lanes 16-31 carry K=2,3
//     -> lane builds {spike(e0), spike(e0+1)} with e0 = eb + 2*(lane>=16)
//   * B layout (4x16 f32): fc value broadcast across all N columns
//     -> every column of D equals I[t]; lane n, VGPR v holds I[t_base + v + 8*(lane>=16)]
// rand loads: lanes 0-15 read 16 consecutive t from row e0 (64B), lanes 16-31
// from row e0+2 (64B) -> fully coalesced, each of the 52.4 MB read once.
// ---------------------------------------------------------------------------
__global__ __launch_bounds__(256) void snn_kernel(const float* __restrict__ rand_m,
                                                  const float* __restrict__ fcw,
                                                  const float* __restrict__ pooled,
                                                  float* __restrict__ out) {
    __shared__ float pool_s[EMB];
    __shared__ float fc_s[EMB];
    __shared__ float Ilds[TSTEP];

    const int b   = blockIdx.x;
    const int tid = threadIdx.x;
    if (tid < EMB) {
        pool_s[tid] = pooled[b * EMB + tid];
        fc_s[tid]   = fcw[tid];
    }
    __syncthreads();

    const int lane    = tid & 31;
    const int wave    = tid >> 5;
    const int halfsel = lane >> 4;      // 0: lanes 0-15, 1: lanes 16-31
    const int mrow    = lane & 15;      // M index within the 16x16 tile
    const float* rb   = rand_m + (size_t)b * EMB * TSTEP;

    #pragma unroll
    for (int j = 0; j < 2; ++j) {       // two 16-timestep tiles per wave
        const int t_base = wave * 32 + j * 16;
        const int tt     = t_base + mrow;
        v8f acc = {};
        for (int eb = 0; eb < EMB; eb += 4) {
            const int e0 = eb + 2 * halfsel;
            const float r0 = rb[(size_t)e0 * TSTEP + tt];
            const float r1 = rb[(size_t)(e0 + 1) * TSTEP + tt];
            v2f a, bm;
            a.x  = pool_s[e0]     > r0 ? 1.0f : 0.0f;   // Bernoulli rate coding
            a.y  = pool_s[e0 + 1] > r1 ? 1.0f : 0.0f;
            bm.x = fc_s[e0];
            bm.y = fc_s[e0 + 1];
            acc = __builtin_amdgcn_wmma_f32_16x16x4_f32(
                /*neg_a=*/false, a, /*neg_b=*/false, bm,
                /*c_mod=*/(short)0, acc, /*reuse_a=*/false, /*reuse_b=*/false);
        }
        // D columns identical; lanes 0 and 16 of each wave own M = v / v+8.
        if (mrow == 0) {
            #pragma unroll
            for (int v = 0; v < 8; ++v)
                Ilds[t_base + v + 8 * halfsel] = acc[v];
        }
    }
    __syncthreads();

    // Sequential LIF (soft reset) over 256 steps; 512 blocks do this in
    // parallel, so the serial chain is off the critical path.
    if (tid == 0) {
        float v  = 0.0f;
        int  cnt = 0;
        for (int t = 0; t < TSTEP; ++t) {
            v = v - (v > THRES ? THRES : 0.0f) + Ilds[t];
            cnt += (v > THRES) ? 1 : 0;
        }
        // EMB * (rate - 0.5) / 2  ==  50 * (cnt/256 - 0.5)
        out[b] = ((float)cnt * (1.0f / (float)TSTEP) - 0.5f) * 50.0f;
    }
}

extern "C" void kernel_launch(void* const* d_in, const int* in_sizes, int n_in,
                              void* d_out, int out_size, void* d_ws, size_t ws_size,
                              hipStream_t stream) {
    const int*   text = (const int*)d_in[0];
    const float* emb  = (const float*)d_in[1];
    const float* fcw  = (const float*)d_in[2];
    const float* rnd  = (const float*)d_in[3];
    float*       out  = (float*)d_out;
    float*       pooled = (float*)d_ws;          // BATCH*EMB floats = 200 KB

    pool_kernel<<<BATCH, 128, 0, stream>>>(text, emb, pooled);
    snn_kernel <<<BATCH, 256, 0, stream>>>(rnd, fcw, pooled, out);
}